// BusStopPredictor_25228637897420
// MI455X (gfx1250) — compile-verified
//
#include <hip/hip_runtime.h>
#include <hip/hip_bf16.h>
#include <math.h>

typedef __attribute__((ext_vector_type(2))) float v2f;
typedef __attribute__((ext_vector_type(8))) float v8f;

#define HID 32

// ---------------- degree / normalization ----------------

__global__ void k_deg_init(float* __restrict__ deg, int n) {
    int i = blockIdx.x * blockDim.x + threadIdx.x;
    if (i < n) deg[i] = 1.0f;  // self-loop
}

__global__ void k_deg_count(const int* __restrict__ dst, float* __restrict__ deg, int e) {
    int i = blockIdx.x * blockDim.x + threadIdx.x;
    if (i < e)
        __hip_atomic_fetch_add(&deg[dst[i]], 1.0f, __ATOMIC_RELAXED, __HIP_MEMORY_SCOPE_AGENT);
}

__global__ void k_deg_fin(float* __restrict__ dinv, int n) {
    int i = blockIdx.x * blockDim.x + threadIdx.x;
    if (i < n) dinv[i] = rsqrtf(dinv[i]);  // deg >= 1 always
}

// ---------------- layer 1: x@W1 (K=2) fused with self-loop accumulator init ----------------
// t1[i,f] = x[i,0]*W1[0,f] + x[i,1]*W1[1,f]
// acc1[i,f] = t1[i,f]*dinv[i]^2 + b1[f]     (self-loop contribution + bias)

__global__ void k_lin1(const float* __restrict__ x, const float* __restrict__ W1,
                       const float* __restrict__ b1, const float* __restrict__ dinv,
                       float* __restrict__ t1, float* __restrict__ acc1, int n) {
    int gid = blockIdx.x * blockDim.x + threadIdx.x;
    int i = gid >> 3;      // node
    int q = gid & 7;       // quad of features
    if (i >= n) return;
    float x0 = x[i * 2 + 0];
    float x1 = x[i * 2 + 1];
    float di = dinv[i];
    float s  = di * di;
    int f = q * 4;
    float4 w0 = *(const float4*)(W1 + f);        // W1[0, f..f+3]
    float4 w1 = *(const float4*)(W1 + HID + f);  // W1[1, f..f+3]
    float4 bb = *(const float4*)(b1 + f);
    float4 t;
    t.x = fmaf(x0, w0.x, x1 * w1.x);
    t.y = fmaf(x0, w0.y, x1 * w1.y);
    t.z = fmaf(x0, w0.z, x1 * w1.z);
    t.w = fmaf(x0, w0.w, x1 * w1.w);
    *(float4*)(t1 + (size_t)i * HID + f) = t;
    float4 a;
    a.x = fmaf(t.x, s, bb.x);
    a.y = fmaf(t.y, s, bb.y);
    a.z = fmaf(t.z, s, bb.z);
    a.w = fmaf(t.w, s, bb.w);
    *(float4*)(acc1 + (size_t)i * HID + f) = a;
}

// ---------------- edge aggregation: acc[dst] += t[src] * dinv[src]*dinv[dst] ----------------

__global__ void k_agg(const float* __restrict__ t, const float* __restrict__ dinv,
                      const int* __restrict__ src, const int* __restrict__ dst,
                      float* __restrict__ acc, int e) {
    int i = blockIdx.x * blockDim.x + threadIdx.x;
    if (i >= e) return;
    int s = src[i];
    int d = dst[i];
    float norm = dinv[s] * dinv[d];
    const float4* tr = (const float4*)(t + (size_t)s * HID);
    float* ar = acc + (size_t)d * HID;
#pragma unroll
    for (int q = 0; q < 8; ++q) {
        float4 v = tr[q];
        __hip_atomic_fetch_add(ar + q * 4 + 0, v.x * norm, __ATOMIC_RELAXED, __HIP_MEMORY_SCOPE_AGENT);
        __hip_atomic_fetch_add(ar + q * 4 + 1, v.y * norm, __ATOMIC_RELAXED, __HIP_MEMORY_SCOPE_AGENT);
        __hip_atomic_fetch_add(ar + q * 4 + 2, v.z * norm, __ATOMIC_RELAXED, __HIP_MEMORY_SCOPE_AGENT);
        __hip_atomic_fetch_add(ar + q * 4 + 3, v.w * norm, __ATOMIC_RELAXED, __HIP_MEMORY_SCOPE_AGENT);
    }
}

// ---------------- layer 2: t2 = relu(a1) @ W2 via V_WMMA_F32_16X16X4_F32 ----------------
// One wave per 16-row tile; two N=16 accumulators; 8 chained K=4 WMMAs each.
// Fused: relu on A load; acc2[i,f] = t2[i,f]*dinv[i]^2 + b2[f] on store.
// ISA layouts (CDNA5 ISA):
//   A 16x4: lanes0-15 M=0..15, v0=K0 v1=K1; lanes16-31 same M, v0=K2 v1=K3
//   B 4x16: lanes0-15 N=0..15, v0=K0 v1=K1; lanes16-31 same N, v0=K2 v1=K3
//   C/D 16x16 f32: vj = row (8*half + j), lane&15 = col

__global__ __launch_bounds__(256) void k_lin2_wmma(
    const float* __restrict__ a1,   // [N,32] pre-relu layer-1 activations (bufB)
    const float* __restrict__ W2,   // [32,32] row-major
    const float* __restrict__ b2,   // [32]
    const float* __restrict__ dinv, // [N]
    float* __restrict__ t2,         // [N,32] out (bufA)
    float* __restrict__ acc2,       // [N,32] out accumulator init (bufB, rewritten in place)
    int nTiles) {
    int wave = (int)((blockIdx.x * blockDim.x + threadIdx.x) >> 5);
    if (wave >= nTiles) return;            // wave-uniform exit: EXEC stays all-ones
    int lane = threadIdx.x & 31;
    int h = lane >> 4;                     // lane half
    int l = lane & 15;
    int rowA = wave * 16 + l;              // A-matrix row for this lane

    v8f c0 = {};  // cols 0..15
    v8f c1 = {};  // cols 16..31
#pragma unroll
    for (int kk = 0; kk < 8; ++kk) {
        int k0 = 4 * kk + 2 * h;
        // A: relu(a1[rowA, k0..k0+1]) — contiguous pair
        v2f a;
        {
            float2 p = *(const float2*)(a1 + (size_t)rowA * HID + k0);
            a.x = p.x > 0.0f ? p.x : 0.0f;
            a.y = p.y > 0.0f ? p.y : 0.0f;
        }
        // B tiles: b[v] = W2[k0+v, col]  (stride-32 rows -> scalar loads, L0-cached)
        v2f b0, b1v;
        b0.x  = W2[(k0 + 0) * HID + l];
        b0.y  = W2[(k0 + 1) * HID + l];
        b1v.x = W2[(k0 + 0) * HID + 16 + l];
        b1v.y = W2[(k0 + 1) * HID + 16 + l];
        c0 = __builtin_amdgcn_wmma_f32_16x16x4_f32(false, a, false, b0,  (short)0, c0, false, false);
        c1 = __builtin_amdgcn_wmma_f32_16x16x4_f32(false, a, false, b1v, (short)0, c1, false, false);
    }
    // D store: row = wave*16 + 8*h + j, col = nt*16 + l.
    // Reads of a1 (same rows) all precede these writes within this wave, so
    // acc2 aliasing bufB is safe.
#pragma unroll
    for (int j = 0; j < 8; ++j) {
        int r = wave * 16 + 8 * h + j;
        float di = dinv[r];
        float s  = di * di;
        float v0 = c0[j];
        float v1 = c1[j];
        t2[(size_t)r * HID + l]        = v0;
        t2[(size_t)r * HID + 16 + l]   = v1;
        acc2[(size_t)r * HID + l]      = fmaf(v0, s, b2[l]);
        acc2[(size_t)r * HID + 16 + l] = fmaf(v1, s, b2[16 + l]);
    }
}

// ---------------- head: sigmoid(relu(acc2) @ Wp + bp) ----------------

__global__ void k_head(const float* __restrict__ acc2, const float* __restrict__ Wp,
                       const float* __restrict__ bp, float* __restrict__ out, int n) {
    int i = blockIdx.x * blockDim.x + threadIdx.x;
    if (i >= n) return;
    const float4* row = (const float4*)(acc2 + (size_t)i * HID);
    float z = 0.0f;
#pragma unroll
    for (int q = 0; q < 8; ++q) {
        float4 v = row[q];
        float4 w = ((const float4*)Wp)[q];
        z = fmaf(fmaxf(v.x, 0.0f), w.x, z);
        z = fmaf(fmaxf(v.y, 0.0f), w.y, z);
        z = fmaf(fmaxf(v.z, 0.0f), w.z, z);
        z = fmaf(fmaxf(v.w, 0.0f), w.w, z);
    }
    z += bp[0];
    out[i] = 1.0f / (1.0f + expf(-z));
}

// ---------------- launcher ----------------

extern "C" void kernel_launch(void* const* d_in, const int* in_sizes, int n_in,
                              void* d_out, int out_size, void* d_ws, size_t ws_size,
                              hipStream_t stream) {
    const float* x          = (const float*)d_in[0];
    const int*   edge_index = (const int*)d_in[1];
    const float* W1         = (const float*)d_in[2];
    const float* b1         = (const float*)d_in[3];
    const float* W2         = (const float*)d_in[4];
    const float* b2         = (const float*)d_in[5];
    const float* Wp         = (const float*)d_in[6];
    const float* bp         = (const float*)d_in[7];
    float* out = (float*)d_out;

    const int N = in_sizes[0] / 2;   // x is [N,2]
    const int E = in_sizes[1] / 2;   // edge_index is [2,E]
    const int* src = edge_index;
    const int* dst = edge_index + E;

    // workspace: dinv[N] | bufA[N*32] | bufB[N*32]  (offsets are 16B aligned)
    float* dinv = (float*)d_ws;
    float* bufA = dinv + N;
    float* bufB = bufA + (size_t)N * HID;

    const int T = 256;
    int gN  = (N + T - 1) / T;
    int gE  = (E + T - 1) / T;
    int gN8 = (N * 8 + T - 1) / T;

    // 1) symmetric normalization coefficients
    k_deg_init<<<gN, T, 0, stream>>>(dinv, N);
    k_deg_count<<<gE, T, 0, stream>>>(dst, dinv, E);
    k_deg_fin<<<gN, T, 0, stream>>>(dinv, N);

    // 2) layer 1: linear (K=2) + self-loop accumulator init, then edge scatter
    k_lin1<<<gN8, T, 0, stream>>>(x, W1, b1, dinv, bufA, bufB, N);
    k_agg<<<gE, T, 0, stream>>>(bufA, dinv, src, dst, bufB, E);

    // 3) layer 2: fused relu + WMMA f32 GEMM + self-loop init, then edge scatter
    int nTiles = (N + 15) / 16;                 // 9375 for N=150000
    int gT = (nTiles + 7) / 8;                  // 8 waves (tiles) per 256-thread block
    k_lin2_wmma<<<gT, T, 0, stream>>>(bufB, W2, b2, dinv, bufA, bufB, nTiles);
    k_agg<<<gE, T, 0, stream>>>(bufA, dinv, src, dst, bufB, E);

    // 4) head
    k_head<<<gN, T, 0, stream>>>(bufB, Wp, bp, out, N);
}